// DotGatConv_24739011625576
// MI455X (gfx1250) — compile-verified
//
#include <hip/hip_runtime.h>
#include <math.h>

// DotGAT for MI455X (gfx1250, wave32).
// Phases:
//   0) init: zero d_out, zero segment-sum, set segment-max to -inf
//   1) ft = feat @ W^T via V_WMMA_F32_16X16X4_F32 (wave computes 16x32 tile, K=128)
//   2) per-edge per-head dots (wave per edge) + atomic segment-max into m
//   3) e = exp(a - m[dst]); atomic segment-sum into s
//   4) out[dst] += ft[src] * e/s[dst]  (wave per edge, float4 lanes, native f32 atomics)
// ft (25.6MB) stays resident in the 192MB L2, so the edge phases are L2-bound.

typedef __attribute__((ext_vector_type(2))) float v2f;
typedef __attribute__((ext_vector_type(8))) float v8f;

#define IN_FEATS 128
#define HEADS 4
#define DHEAD 32
#define SCALE 0.17677669529663687f  // 1/sqrt(32)

__device__ __forceinline__ float atomAddF32(float* addr, float val) {
    // Guaranteed native GLOBAL_ATOMIC_ADD_F32 on AMD (no CAS loop).
    return unsafeAtomicAdd(addr, val);
}

__device__ __forceinline__ void atomicMaxFloat(float* addr, float val) {
    // sign-split trick: lowers to native integer global atomics.
    if (val >= 0.0f) {
        atomicMax((int*)addr, __float_as_int(val));
    } else {
        atomicMin((unsigned int*)addr, (unsigned int)__float_as_int(val));
    }
}

// ---------------- phase 0: init ----------------
__global__ void init_kernel(float* __restrict__ out, float* __restrict__ m,
                            float* __restrict__ s, int nOut, int nMH) {
    int i = blockIdx.x * blockDim.x + threadIdx.x;
    if (i < nOut) out[i] = 0.0f;
    if (i < nMH) {
        m[i] = __int_as_float(0xFF800000);  // -inf
        s[i] = 0.0f;
    }
}

// ---------------- phase 1: projection GEMM via f32 WMMA ----------------
// One wave computes a 16(M=nodes) x 32(N=out-feature) tile (two accumulators
// sharing the A operand). 8 waves/block: 2 row-strips x 4 col-pairs
// -> block covers 32 rows x 128 cols. Grid = ceil(N/32).
__global__ void gemm_wmma_f32(const float* __restrict__ feat,
                              const float* __restrict__ W,
                              float* __restrict__ ft, int N) {
    const int lane = threadIdx.x & 31;
    const int wave = threadIdx.x >> 5;              // 0..7
    const int row0 = blockIdx.x * 32 + (wave >> 2) * 16;
    const int col0 = (wave & 3) * 32;
    const int half = lane >> 4;                     // K-pair selector
    const int lr   = lane & 15;                     // M for A-lane, N for B-lane

    // clamp row so every lane has a valid address (WMMA needs full EXEC)
    const int arow_i = (row0 + lr < N) ? (row0 + lr) : (N - 1);
    const float* arow  = feat + (size_t)arow_i * IN_FEATS;
    const float* b0row = W + (size_t)(col0 + lr) * IN_FEATS;       // W: [128 out, 128 in]
    const float* b1row = W + (size_t)(col0 + 16 + lr) * IN_FEATS;

    v8f acc0 = {}, acc1 = {};
    #pragma unroll 4
    for (int kb = 0; kb < IN_FEATS; kb += 4) {
        const int k = kb + 2 * half;
        v2f a;  a.x  = arow[k];  a.y  = arow[k + 1];
        v2f b0; b0.x = b0row[k]; b0.y = b0row[k + 1];
        v2f b1; b1.x = b1row[k]; b1.y = b1row[k + 1];
        acc0 = __builtin_amdgcn_wmma_f32_16x16x4_f32(false, a, false, b0,
                                                     (short)0, acc0, false, false);
        acc1 = __builtin_amdgcn_wmma_f32_16x16x4_f32(false, a, false, b1,
                                                     (short)0, acc1, false, false);
    }

    // C/D layout: VGPR r, lanes 0-15 -> M=r, lanes 16-31 -> M=8+r; N = lane%16
    const int mbase = row0 + 8 * half;
    if (row0 + 16 <= N) {
        // fast path: whole tile in bounds, no per-row guards
        float* o = ft + (size_t)mbase * IN_FEATS + col0 + lr;
        #pragma unroll
        for (int r = 0; r < 8; ++r) {
            o[(size_t)r * IN_FEATS]      = acc0[r];
            o[(size_t)r * IN_FEATS + 16] = acc1[r];
        }
    } else {
        #pragma unroll
        for (int r = 0; r < 8; ++r) {
            const int row = mbase + r;
            if (row < N) {
                ft[(size_t)row * IN_FEATS + col0 + lr]      = acc0[r];
                ft[(size_t)row * IN_FEATS + col0 + 16 + lr] = acc1[r];
            }
        }
    }
}

// ---------------- phase 2: edge dots + segment max ----------------
// Wave per edge: lane l loads float4 at feature offset 4*l -> head h = l>>3.
__global__ void edge_dot_kernel(const float* __restrict__ ft,
                                const int* __restrict__ src,
                                const int* __restrict__ dst,
                                float* __restrict__ a,      // [E, 4]
                                float* __restrict__ m,      // [N, 4]
                                int E) {
    const int e = blockIdx.x * 8 + (threadIdx.x >> 5);
    if (e >= E) return;
    const int lane = threadIdx.x & 31;
    const int sn = src[e], dn = dst[e];
    const float4 x = ((const float4*)(ft + (size_t)sn * IN_FEATS))[lane];
    const float4 y = ((const float4*)(ft + (size_t)dn * IN_FEATS))[lane];
    float v = x.x * y.x + x.y * y.y + x.z * y.z + x.w * y.w;
    // reduce the 8 lanes of each head group
    v += __shfl_xor(v, 1, 32);
    v += __shfl_xor(v, 2, 32);
    v += __shfl_xor(v, 4, 32);
    if ((lane & 7) == 0) {
        const int h = lane >> 3;
        const float av = v * SCALE;
        a[(size_t)e * HEADS + h] = av;
        atomicMaxFloat(m + (size_t)dn * HEADS + h, av);
    }
}

// ---------------- phase 3: exp + segment sum ----------------
__global__ void edge_exp_kernel(const int* __restrict__ dst,
                                float* __restrict__ a,      // in: logits, out: exp
                                const float* __restrict__ m,
                                float* __restrict__ s,
                                int EH) {
    const int i = blockIdx.x * blockDim.x + threadIdx.x;
    if (i >= EH) return;
    const int e = i >> 2;
    const int h = i & 3;
    const int dn = dst[e];
    const float ex = __expf(a[i] - m[(size_t)dn * HEADS + h]);
    a[i] = ex;
    atomAddF32(s + (size_t)dn * HEADS + h, ex);
}

// ---------------- phase 4: weighted scatter-add aggregation ----------------
__global__ void aggregate_kernel(const float* __restrict__ ft,
                                 const int* __restrict__ src,
                                 const int* __restrict__ dst,
                                 const float* __restrict__ en,  // exp numerators [E,4]
                                 const float* __restrict__ s,   // segment sums [N,4]
                                 float* __restrict__ out, int E) {
    const int e = blockIdx.x * 8 + (threadIdx.x >> 5);
    if (e >= E) return;
    const int lane = threadIdx.x & 31;
    const int sn = src[e], dn = dst[e];
    const int h = lane >> 3;
    const float w = en[(size_t)e * HEADS + h] / s[(size_t)dn * HEADS + h];
    const float4 x = ((const float4*)(ft + (size_t)sn * IN_FEATS))[lane];
    float* o = out + (size_t)dn * IN_FEATS + lane * 4;
    atomAddF32(o + 0, x.x * w);
    atomAddF32(o + 1, x.y * w);
    atomAddF32(o + 2, x.z * w);
    atomAddF32(o + 3, x.w * w);
}

extern "C" void kernel_launch(void* const* d_in, const int* in_sizes, int n_in,
                              void* d_out, int out_size, void* d_ws, size_t ws_size,
                              hipStream_t stream) {
    const float* feat = (const float*)d_in[0];   // [N, 128]
    const float* W    = (const float*)d_in[1];   // [128, 128]
    const int*   src  = (const int*)d_in[2];     // [E]
    const int*   dst  = (const int*)d_in[3];     // [E]
    float* out = (float*)d_out;                  // [N, 4, 32]

    const int N = in_sizes[0] / IN_FEATS;
    const int E = in_sizes[2];

    // workspace layout (floats): ft[N*128] | a[E*4] | m[N*4] | s[N*4]
    float* ft = (float*)d_ws;
    float* a  = ft + (size_t)N * IN_FEATS;
    float* m  = a  + (size_t)E * HEADS;
    float* s  = m  + (size_t)N * HEADS;

    const int nOut = N * IN_FEATS;
    const int nMH  = N * HEADS;

    init_kernel<<<(nOut + 255) / 256, 256, 0, stream>>>(out, m, s, nOut, nMH);

    gemm_wmma_f32<<<(N + 31) / 32, 256, 0, stream>>>(feat, W, ft, N);

    edge_dot_kernel<<<(E + 7) / 8, 256, 0, stream>>>(ft, src, dst, a, m, E);

    const int EH = E * HEADS;
    edge_exp_kernel<<<(EH + 255) / 256, 256, 0, stream>>>(dst, a, m, s, EH);

    aggregate_kernel<<<(E + 7) / 8, 256, 0, stream>>>(ft, src, dst, a, s, out, E);
}